// InferenceModel_41807211660012
// MI455X (gfx1250) — compile-verified
//
#include <hip/hip_runtime.h>
#include <hip/hip_bf16.h>

typedef float v2f __attribute__((ext_vector_type(2)));
typedef float v8f __attribute__((ext_vector_type(8)));

#define EPS_F 1e-12f
#define LN2_D 0.6931471805599453094

// Each wave handles 8 consecutive 16x16 (i,j)-tiles of one i-tile row.
// Dot products y_i . y_j for a whole 16x16 tile come from one
// V_WMMA_F32_16X16X4_F32 (K=3 padded to 4 with zeros).
// NC > 0: compile-time n (constant-folds all Pstar load offsets into
// immediate IOFFSETs off a single per-group base address).
template<int NC>
__global__ void __launch_bounds__(256) bce_q_wmma_kernel(
    const float* __restrict__ Pstar,   // [n, n]
    const float* __restrict__ pp,      // [n, 3]
    double* __restrict__ acc_out,      // [1] f64 accumulator (pre-zeroed)
    int n_rt)
{
    const int n = (NC > 0) ? NC : n_rt;

    const int GRP = (n >> 4) >> 3;           // groups of 8 j-tiles per i-row
    const int G   = (n >> 4) * GRP;          // total wave work-groups

    const int lane = threadIdx.x & 31;
    const int lo   = lane & 15;
    const int hi   = lane >> 4;              // 0 or 1 (wave32 halves)
    const int wid     = blockIdx.x * (blockDim.x >> 5) + (threadIdx.x >> 5);
    const int wstride = gridDim.x * (blockDim.x >> 5);

    float acc0 = 0.0f, acc1 = 0.0f;  // log2-domain partial sums (x ln2 at end)

    for (int g = wid; g < G; g += wstride) {        // wave-uniform loop
        const int i0     = (g / GRP) << 4;
        const int j0base = (g % GRP) << 7;

        // ---- A operand: rows i0..i0+15 of y, 16x4 f32 (K=3, pad K=3 with 0)
        // Layout: lane holds M = lane&15; K = vgpr + 2*hi.
        const int i = i0 + lo;
        const float yi0 = pp[3*i+0], yi1 = pp[3*i+1], yi2 = pp[3*i+2];
        const float sq_self = yi0*yi0 + yi1*yi1 + yi2*yi2;
        v2f a;
        a.x = hi ? yi2 : yi0;
        a.y = hi ? 0.0f : yi1;

        // sq[i0 + r + 8*hi]: lane s holds sq of row i0+(s&15) -> shuffle
        float sqi[8];
#pragma unroll
        for (int r = 0; r < 8; ++r)
            sqi[r] = __shfl(sq_self, r + 8*hi, 32);

        // ---- Preload B operands for all 8 j-tiles (constant offsets 192*t B)
        float bx[8], by[8], sqj[8];
        const float* jbase = pp + 3 * (size_t)(j0base + lo);
#pragma unroll
        for (int t = 0; t < 8; ++t) {
            const float yj0 = jbase[48*t + 0];
            const float yj1 = jbase[48*t + 1];
            const float yj2 = jbase[48*t + 2];
            sqj[t] = yj0*yj0 + yj1*yj1 + yj2*yj2;
            bx[t]  = hi ? yj2 : yj0;
            by[t]  = hi ? 0.0f : yj1;
        }

        // Single base address for all 64 Pstar loads of this group:
        // element (r, t) lives at pbase[r*n + 16*t]  (immediate offsets for NC>0)
        const float* pbase = Pstar + (size_t)(i0 + 8*hi) * (size_t)n
                                   + (size_t)(j0base + lo);

#pragma unroll
        for (int t = 0; t < 8; ++t) {
            // Batch-issue the 8 P loads for this tile before the WMMA so the
            // matrix op + transcendental math overlap the memory latency.
            float pv[8];
#pragma unroll
            for (int r = 0; r < 8; ++r)
                pv[r] = pbase[(size_t)r * (size_t)n + 16*t];

            v2f b; b.x = bx[t]; b.y = by[t];
            v8f c = {0.f, 0.f, 0.f, 0.f, 0.f, 0.f, 0.f, 0.f};
            // D = A x B: c[r] = dot(y[i0+r+8*hi], y[j0base+16t+lo])
            c = __builtin_amdgcn_wmma_f32_16x16x4_f32(
                    false, a, false, b, (short)0, c, false, false);

#pragma unroll
            for (int r = 0; r < 8; ++r) {
                // clip(p, 1e-12, 1-1e-12): one v_med3_f32 (1-1e-12 == 1.0f)
                const float p = __builtin_amdgcn_fmed3f(pv[r], EPS_F, 1.0f);
                const float omp = 1.0f - p;
                float d2 = (sqi[r] + sqj[t]) - 2.0f * c[r];
                d2 = fmaxf(d2, 0.0f);
                const float q   = __builtin_amdgcn_rcpf(1.0f + d2);
                const float Lp  = __builtin_amdgcn_logf(p);    // log2(p)
                const float Lom = __builtin_amdgcn_logf(omp);  // log2(1-p)
                const float term = Lom + q * (Lp - Lom);       // log2 domain
                if (r & 1) acc1 += term; else acc0 += term;
            }
        }
    }

    float acc = acc0 + acc1;

    // wave32 reduction
#pragma unroll
    for (int off = 16; off > 0; off >>= 1)
        acc += __shfl_xor(acc, off, 32);

    __shared__ float wsum[8];
    if (lane == 0) wsum[threadIdx.x >> 5] = acc;
    __syncthreads();
    if (threadIdx.x == 0) {
        float bsum = 0.0f;
        const int nw = blockDim.x >> 5;
        for (int w = 0; w < nw; ++w) bsum += wsum[w];
        atomicAdd(acc_out, (double)bsum);            // global_atomic_add_f64
    }
}

__global__ void init_ws_kernel(double* __restrict__ ws) { ws[0] = 0.0; }

__global__ void finalize_kernel(const double* __restrict__ ws,
                                const float* __restrict__ pp,
                                float* __restrict__ out,
                                int n, int m /* = 3n */)
{
    const int idx = blockIdx.x * blockDim.x + threadIdx.x;
    if (idx == 0) {
        const double mean = ws[0] * LN2_D / ((double)n * (double)n);
        out[0] = (float)(-mean);
    }
    if (idx >= 1 && idx <= m) out[idx] = pp[idx - 1];
}

extern "C" void kernel_launch(void* const* d_in, const int* in_sizes, int n_in,
                              void* d_out, int out_size, void* d_ws, size_t ws_size,
                              hipStream_t stream) {
    const float* Pstar = (const float*)d_in[0];   // [n, n] f32
    const float* pp    = (const float*)d_in[1];   // [n, 3] f32
    const int n = in_sizes[1] / 3;                // 8192

    double* ws  = (double*)d_ws;
    float*  out = (float*)d_out;

    init_ws_kernel<<<1, 1, 0, stream>>>(ws);

    // total wave work-groups: (n/16 i-tiles) * (n/128 j-groups)
    const int G = (n >> 4) * ((n >> 4) >> 3);     // 32768 for n=8192
    const int wavesPerBlock = 8;                  // 256 threads, wave32
    int blocks = (G + wavesPerBlock - 1) / wavesPerBlock;  // 4096
    if (blocks < 1) blocks = 1;

    if (n == 8192) {
        bce_q_wmma_kernel<8192><<<blocks, 256, 0, stream>>>(Pstar, pp, ws, n);
    } else {
        bce_q_wmma_kernel<0><<<blocks, 256, 0, stream>>>(Pstar, pp, ws, n);
    }

    const int m   = 3 * n;
    const int tot = m + 1;                        // out_size
    finalize_kernel<<<(tot + 255) / 256, 256, 0, stream>>>(ws, pp, out, n, m);
}